// Group_74509092651098
// MI455X (gfx1250) — compile-verified
//
#include <hip/hip_runtime.h>

#define BATCH 16
#define NPTS  16384
#define NGRP  512
#define KNN_K 32
#define BIGF  1e10f

#define CHUNK    2048
#define D2STRIDE 2052   // pad so rows m and m+8 hit disjoint LDS banks

typedef float v2f __attribute__((ext_vector_type(2)));
typedef float v8f __attribute__((ext_vector_type(8)));

// -------------------- Kernel 1: farthest point sampling --------------------
// One workgroup per batch; all points resident in LDS (256KB of the 320KB WGP LDS).
__global__ __launch_bounds__(1024) void fps_kernel(const float* __restrict__ xyz,
                                                   float* __restrict__ centerOut) {
  __shared__ float4 pts[NPTS];     // 256 KB
  __shared__ float  redV[32];
  __shared__ int    redI[32];
  __shared__ int    curSel;

  const int b    = blockIdx.x;
  const int t    = threadIdx.x;
  const int lane = t & 31;
  const int wave = t >> 5;
  const float* xb = xyz + (size_t)b * NPTS * 3;

  for (int i = t; i < NPTS; i += 1024) {
    float x = xb[i * 3 + 0], y = xb[i * 3 + 1], z = xb[i * 3 + 2];
    pts[i] = make_float4(x, y, z, 0.0f);
  }
  if (t == 0) curSel = 0;
  __syncthreads();

  float mind[16];
#pragma unroll
  for (int j = 0; j < 16; ++j) mind[j] = BIGF;

  int cur = 0;
  for (int it = 0; it < NGRP; ++it) {
    float4 c = pts[cur];                      // broadcast read
    if (t == 0) {                             // record center for this selection
      size_t o = ((size_t)b * NGRP + it) * 3;
      centerOut[o + 0] = c.x;
      centerOut[o + 1] = c.y;
      centerOut[o + 2] = c.z;
    }
    float bv = -1.0f;
    int   bi = 0x7fffffff;
#pragma unroll
    for (int j = 0; j < 16; ++j) {
      int p = t + j * 1024;                   // strided ownership: conflict-free b128 loads
      float4 q = pts[p];
      float dx = q.x - c.x, dy = q.y - c.y, dz = q.z - c.z;
      float d2 = dx * dx + dy * dy + dz * dz;
      float m = fminf(mind[j], d2);
      mind[j] = m;
      if (m > bv) { bv = m; bi = p; }         // ascending p -> keeps lowest tied index
    }
    // wave32 reduce: max value, lowest index on tie (matches jnp.argmax)
#pragma unroll
    for (int off = 16; off; off >>= 1) {
      float ov = __shfl_xor(bv, off);
      int   oi = __shfl_xor(bi, off);
      if (ov > bv || (ov == bv && oi < bi)) { bv = ov; bi = oi; }
    }
    if (lane == 0) { redV[wave] = bv; redI[wave] = bi; }
    __syncthreads();
    if (wave == 0) {
      bv = redV[lane]; bi = redI[lane];
#pragma unroll
      for (int off = 16; off; off >>= 1) {
        float ov = __shfl_xor(bv, off);
        int   oi = __shfl_xor(bi, off);
        if (ov > bv || (ov == bv && oi < bi)) { bv = ov; bi = oi; }
      }
      if (lane == 0) curSel = bi;
    }
    __syncthreads();
    cur = curSel;
  }
}

// -------------------- Kernel 2: KNN via WMMA + gather --------------------
// grid = (NGRP/16, BATCH). Each WG: 16 centers, 32 waves.
// d2rel = x^2 - 2*c.x  (drops per-row constant c^2: same top-k ordering).
// WMMA: A[16x4] = (-2cx,-2cy,-2cz,1) per center, B[4x16] = (x,y,z,x^2) per point.
__global__ __launch_bounds__(1024) void knn_kernel(const float* __restrict__ xyz,
                                                   const float* __restrict__ centerIn,
                                                   float* __restrict__ nbrOut,
                                                   int* __restrict__ idxOut) {
  __shared__ float4 pts[CHUNK];               // 32 KB staged chunk (x,y,z,x^2)
  __shared__ float  d2s[16 * D2STRIDE];       // 128 KB distance tile

  const int gt   = blockIdx.x;                // group tile 0..31
  const int b    = blockIdx.y;                // batch
  const int t    = threadIdx.x;
  const int lane = t & 31;
  const int wave = t >> 5;
  const int g0   = gt * 16;

  const float* xb = xyz + (size_t)b * NPTS * 3;
  const float* cb = centerIn + ((size_t)b * NGRP + g0) * 3;

  // Build A-matrix fragment (ISA 16x4 f32 layout: lanes 0-15 hold K=0,1; lanes 16-31 hold K=2,3)
  const int am = lane & 15;
  const float cx = cb[am * 3 + 0], cy = cb[am * 3 + 1], cz = cb[am * 3 + 2];
  v2f A;
  A.x = (lane < 16) ? (-2.0f * cx) : (-2.0f * cz);
  A.y = (lane < 16) ? (-2.0f * cy) : 1.0f;

  // Register-resident sorted top-K: waves 0..15 own center row == wave, lane l = slot l.
  float cd = BIGF;
  int   ci = 0x7fffffff;

  for (int ch = 0; ch < NPTS / CHUNK; ++ch) {
    const int pbase = ch * CHUNK;
    for (int i = t; i < CHUNK; i += 1024) {
      const float* p = xb + (size_t)(pbase + i) * 3;
      float x = p[0], y = p[1], z = p[2];
      pts[i] = make_float4(x, y, z, x * x + y * y + z * z);
    }
    if (ch + 1 < NPTS / CHUNK)                // warm L2/L0 for next chunk
      __builtin_prefetch(xb + (size_t)(pbase + CHUNK) * 3 + (size_t)t * 2, 0, 0);
    __syncthreads();

    // Each wave: 64 points = 4 WMMA tiles of 16x16.
#pragma unroll
    for (int tile = 0; tile < 4; ++tile) {
      const int pb = wave * 64 + tile * 16;
      float4 q = pts[pb + (lane & 15)];
      v2f Bv;                                 // B 4x16: lanes0-15 K=0,1 ; lanes16-31 K=2,3
      Bv.x = (lane < 16) ? q.x : q.z;
      Bv.y = (lane < 16) ? q.y : q.w;
      v8f C = {};
      v8f D = __builtin_amdgcn_wmma_f32_16x16x4_f32(false, A, false, Bv,
                                                    (short)0, C, false, false);
      const int row0 = (lane >> 4) * 8;       // D layout: vgpr v -> row v + 8*(lane/16)
      const int col  = pb + (lane & 15);
#pragma unroll
      for (int v = 0; v < 8; ++v)
        d2s[(row0 + v) * D2STRIDE + col] = D[v];
    }
    __syncthreads();

    // Merge chunk into running sorted top-32 (lexicographic (d, idx) = top_k semantics).
    if (wave < 16) {
      float td = __shfl(cd, 31);
      int   ti = __shfl(ci, 31);
      for (int c0 = 0; c0 < CHUNK; c0 += 32) {
        float dv = d2s[wave * D2STRIDE + c0 + lane];
        int   gi = pbase + c0 + lane;
        bool cand = (dv < td) || (dv == td && gi < ti);
        unsigned long long mask = __ballot(cand);
        while (mask) {
          int src = __ffsll((long long)mask) - 1;   // lowest lane first -> lowest index first
          mask &= mask - 1;
          float nd = __shfl(dv, src);
          int   ni = __shfl(gi, src);
          if (nd < td || (nd == td && ni < ti)) {
            bool lt = (cd < nd) || (cd == nd && ci < ni);
            int pos = __popcll(__ballot(lt));       // insertion position
            float pd = __shfl_up(cd, 1);
            int   pi = __shfl_up(ci, 1);
            if (lane > pos)       { cd = pd; ci = pi; }
            else if (lane == pos) { cd = nd; ci = ni; }
            td = __shfl(cd, 31);
            ti = __shfl(ci, 31);
          }
        }
      }
    }
    __syncthreads();
  }

  // Emit idx + gathered, centered neighborhood.
  if (wave < 16) {
    const int g = g0 + wave;
    idxOut[((size_t)b * NGRP + g) * KNN_K + lane] = ci;
    const float gcx = cb[wave * 3 + 0], gcy = cb[wave * 3 + 1], gcz = cb[wave * 3 + 2];
    const float* p = xb + (size_t)ci * 3;
    float* o = nbrOut + (((size_t)b * NGRP + g) * KNN_K + lane) * 3;
    o[0] = p[0] - gcx;
    o[1] = p[1] - gcy;
    o[2] = p[2] - gcz;
  }
}

extern "C" void kernel_launch(void* const* d_in, const int* in_sizes, int n_in,
                              void* d_out, int out_size, void* d_ws, size_t ws_size,
                              hipStream_t stream) {
  (void)in_sizes; (void)n_in; (void)d_ws; (void)ws_size; (void)out_size;
  const float* xyz = (const float*)d_in[0];

  // d_out = [neighborhood (B,G,K,3) | center (B,G,3) | idx (B,G,K) as int32]
  float* nbrOut    = (float*)d_out;
  float* centerOut = nbrOut + (size_t)BATCH * NGRP * KNN_K * 3;
  int*   idxOut    = (int*)(centerOut + (size_t)BATCH * NGRP * 3);

  fps_kernel<<<dim3(BATCH), dim3(1024), 0, stream>>>(xyz, centerOut);
  knn_kernel<<<dim3(NGRP / 16, BATCH), dim3(1024), 0, stream>>>(xyz, centerOut,
                                                                nbrOut, idxOut);
}